// VolGRUCellTorch_90915867721909
// MI455X (gfx1250) — compile-verified
//
#include <hip/hip_runtime.h>
#include <math.h>

// N = 262144 timesteps, F = 32 features, S = 32 state dims, H = 64 hidden.

typedef float v2f __attribute__((ext_vector_type(2)));
typedef float v8f __attribute__((ext_vector_type(8)));

// ---------------------------------------------------------------------------
// Kernel 1 (unchanged from round 1 — verified clean WMMA codegen):
// GL = X @ Bg, RL = X @ Br with exact-f32 V_WMMA_F32_16X16X4_F32.
// ---------------------------------------------------------------------------
__global__ __launch_bounds__(256) void volgru_gemm_wmma(
    const float* __restrict__ X,
    const float* __restrict__ Bg,
    const float* __restrict__ Br,
    float* __restrict__ GL,
    float* __restrict__ RL,
    int n)
{
    const int lane = threadIdx.x & 31;
    const int wave = blockIdx.x * (blockDim.x >> 5) + (threadIdx.x >> 5);
    const int row0 = wave << 4;
    if (row0 + 16 > n) return;             // wave-uniform guard (EXEC stays all-ones)

    const int half = lane >> 4;
    const int l16  = lane & 15;

    v8f aG0 = {}, aG1 = {}, aR0 = {}, aR1 = {};

    #pragma unroll
    for (int k0 = 0; k0 < 32; k0 += 4) {
        const int kr = k0 + half * 2;
        v2f a;
        const float* ap = X + (size_t)(row0 + l16) * 32 + kr;
        a.x = ap[0];
        a.y = ap[1];

        v2f bg0, bg1, br0, br1;
        bg0.x = Bg[kr * 32 + l16];       bg0.y = Bg[(kr + 1) * 32 + l16];
        bg1.x = Bg[kr * 32 + 16 + l16];  bg1.y = Bg[(kr + 1) * 32 + 16 + l16];
        br0.x = Br[kr * 32 + l16];       br0.y = Br[(kr + 1) * 32 + l16];
        br1.x = Br[kr * 32 + 16 + l16];  br1.y = Br[(kr + 1) * 32 + 16 + l16];

        aG0 = __builtin_amdgcn_wmma_f32_16x16x4_f32(false, a, false, bg0, (short)0, aG0, false, false);
        aG1 = __builtin_amdgcn_wmma_f32_16x16x4_f32(false, a, false, bg1, (short)0, aG1, false, false);
        aR0 = __builtin_amdgcn_wmma_f32_16x16x4_f32(false, a, false, br0, (short)0, aR0, false, false);
        aR1 = __builtin_amdgcn_wmma_f32_16x16x4_f32(false, a, false, br1, (short)0, aR1, false, false);
    }

    #pragma unroll
    for (int r = 0; r < 8; r++) {
        const int row = row0 + r + half * 8;
        GL[(size_t)row * 32 + l16]      = aG0[r];
        GL[(size_t)row * 32 + 16 + l16] = aG1[r];
        RL[(size_t)row * 32 + l16]      = aR0[r];
        RL[(size_t)row * 32 + 16 + l16] = aR1[r];
    }
}

// ---------------------------------------------------------------------------
// Cross-lane broadcast with a compile-time lane index: v_readlane_b32.
// Result lands in an SGPR and feeds v_fmac as the scalar operand — no DS
// hardware, no DScnt waits (round-1 version paid 101 ds_bpermute + waits
// per timestep).
// ---------------------------------------------------------------------------
__device__ __forceinline__ float rdlane(float x, int i) {
#if __has_builtin(__builtin_amdgcn_readlane)
    return __int_as_float(__builtin_amdgcn_readlane(__float_as_int(x), i));
#else
    return __shfl(x, i, 32);
#endif
}

__device__ __forceinline__ float fast_rcp(float x) {
#if __has_builtin(__builtin_amdgcn_rcpf)
    return __builtin_amdgcn_rcpf(x);
#else
    return 1.0f / x;
#endif
}

__device__ __forceinline__ float fast_tanh(float x) {
#if __has_builtin(__builtin_amdgcn_tanhf)
    return __builtin_amdgcn_tanhf(x);     // V_TANH_F32 (gfx1250 TRANS op)
#else
    return tanhf(x);
#endif
}

__device__ __forceinline__ float dev_sigmoid(float x) {
    return fast_rcp(1.0f + __expf(-x));
}

__device__ __forceinline__ float dev_softplus(float x) {
    // log1p(exp(x)) with range splits: exact-ish and branch-lite
    const float e = __expf(x);
    float sp = __logf(1.0f + e);
    sp = (x < -8.0f) ? e : sp;            // 1+e rounds to 1 below this; use e
    sp = (x > 20.0f) ? x : sp;
    return sp;
}

// ---------------------------------------------------------------------------
// Kernel 2: the nonlinear recurrence. One wave32, lane s == state dim s.
// Weights register-resident; W1 stage packed as v2f to coax V_PK_FMA_F32
// (lane j computes hidden rows j and j+32 simultaneously); W2 stage uses 4
// partial accumulators (chain depth 16). Zero LDS/DS usage.
// ---------------------------------------------------------------------------
__global__ __launch_bounds__(32) void volgru_scan(
    const float* __restrict__ GL, const float* __restrict__ RL,
    const float* __restrict__ X,
    const float* __restrict__ rets,
    const float* __restrict__ Bg,  const float* __restrict__ gsw,
    const float* __restrict__ Br,  const float* __restrict__ rsw,
    const float* __restrict__ W1,  const float* __restrict__ b1,
    const float* __restrict__ W2,  const float* __restrict__ b2,
    float* __restrict__ out_sigma, float* __restrict__ out_z,
    float* __restrict__ out_cand,  float* __restrict__ out_vfin,
    int n, int use_ws)
{
    const int lane = threadIdx.x;            // 0..31

    // W1 rows (lane, lane+32) interleaved as v2f pairs -> packed-FMA friendly
    v2f w1p[35];
    #pragma unroll
    for (int i = 0; i < 35; ++i) {
        w1p[i].x = W1[lane * 35 + i];
        w1p[i].y = W1[(lane + 32) * 35 + i];
    }
    float w2r[64];
    #pragma unroll
    for (int j = 0; j < 64; ++j) w2r[j] = W2[lane * 64 + j];

    v2f b1p;
    b1p.x = b1[lane];
    b1p.y = b1[lane + 32];
    const float b2s   = b2[lane];
    const float gsw_s = gsw[lane], rsw_s = rsw[lane];

    const float r0 = rets[0];
    float v = r0 * r0;                       // init_var=None path

    for (int t = 0; t < n; ++t) {
        if (use_ws && (t + 64) < n) {
            __builtin_prefetch((const void*)(GL + (size_t)(t + 64) * 32 + lane), 0, 0);
            __builtin_prefetch((const void*)(RL + (size_t)(t + 64) * 32 + lane), 0, 0);
        }

        float gl, rl;
        if (use_ws) {
            gl = GL[(size_t)t * 32 + lane];
            rl = RL[(size_t)t * 32 + lane];
        } else {                              // fallback: recompute projections inline
            gl = 0.0f; rl = 0.0f;
            const float* xr = X + (size_t)t * 32;
            #pragma unroll 8
            for (int i = 0; i < 32; ++i) {
                const float xi = xr[i];       // wave-uniform load
                gl = fmaf(xi, Bg[i * 32 + lane], gl);
                rl = fmaf(xi, Br[i * 32 + lane], rl);
            }
        }

        const float r   = rets[t];
        const float z   = dev_sigmoid(fmaf(gsw_s, v, gl));
        const float rst = dev_sigmoid(fmaf(rsw_s, v, rl));
        const float rv  = rst * v;
        const float ar  = fabsf(r), r2 = r * r;

        // ---- h = tanh(W1 @ [r,|r|,r^2, rv] + b1), rows (j, j+32) packed ----
        v2f p0 = b1p, p1 = {0.0f, 0.0f};
        {
            v2f s0; s0.x = r;  s0.y = r;
            v2f s1; s1.x = ar; s1.y = ar;
            v2f s2; s2.x = r2; s2.y = r2;
            p0 += w1p[0] * s0;
            p1 += w1p[1] * s1;
            p0 += w1p[2] * s2;
        }
        #pragma unroll
        for (int i = 0; i < 32; i += 2) {
            const float xb0 = rdlane(rv, i);
            const float xb1 = rdlane(rv, i + 1);
            v2f x0; x0.x = xb0; x0.y = xb0;
            v2f x1; x1.x = xb1; x1.y = xb1;
            p0 += w1p[3 + i]     * x0;
            p1 += w1p[3 + i + 1] * x1;
        }
        const v2f a01 = p0 + p1;
        const float h0 = fast_tanh(a01.x);
        const float h1 = fast_tanh(a01.y);

        // ---- raw = W2 @ h + b2; 4 partial accumulators, readlane bcasts ----
        float q0 = b2s, q1 = 0.0f, q2 = 0.0f, q3 = 0.0f;
        #pragma unroll
        for (int j = 0; j < 16; ++j) {
            q0 = fmaf(w2r[j],      rdlane(h0, j),      q0);
            q1 = fmaf(w2r[16 + j], rdlane(h0, 16 + j), q1);
            q2 = fmaf(w2r[32 + j], rdlane(h1, j),      q2);
            q3 = fmaf(w2r[48 + j], rdlane(h1, 16 + j), q3);
        }
        const float raw  = (q0 + q1) + (q2 + q3);
        const float cand = dev_softplus(raw);
        const float vn   = fmaxf(fmaf(z, cand - v, v), 1e-12f);

        out_z[(size_t)t * 32 + lane]    = z;
        out_cand[(size_t)t * 32 + lane] = cand;

        // mean(v_next): readlane accumulation (uniform result, zero DS traffic)
        float m = 0.0f;
        #pragma unroll
        for (int i = 0; i < 32; ++i) m += rdlane(vn, i);
        if (lane == 0) out_sigma[t] = m * (1.0f / 32.0f);

        v = vn;
    }
    out_vfin[lane] = v;
}

// ---------------------------------------------------------------------------
// Host launcher
// ---------------------------------------------------------------------------
extern "C" void kernel_launch(void* const* d_in, const int* in_sizes, int n_in,
                              void* d_out, int out_size, void* d_ws, size_t ws_size,
                              hipStream_t stream)
{
    const float* X    = (const float*)d_in[0];
    const float* rets = (const float*)d_in[1];
    const float* Bg   = (const float*)d_in[2];
    const float* gsw  = (const float*)d_in[3];
    const float* Br   = (const float*)d_in[4];
    const float* rsw  = (const float*)d_in[5];
    const float* W1   = (const float*)d_in[6];
    const float* b1   = (const float*)d_in[7];
    const float* W2   = (const float*)d_in[8];
    const float* b2   = (const float*)d_in[9];

    const int n = in_sizes[1];               // sequence length (262144)

    float* out       = (float*)d_out;        // [sigma2(n) | z(n*32) | cand(n*32) | v_final(32)]
    float* out_sigma = out;
    float* out_z     = out + (size_t)n;
    float* out_cand  = out_z + (size_t)n * 32;
    float* out_vfin  = out_cand + (size_t)n * 32;

    const size_t need = 2ull * (size_t)n * 32 * sizeof(float);
    const int use_ws = (ws_size >= need) ? 1 : 0;
    float* GL = (float*)d_ws;
    float* RL = GL + (size_t)n * 32;

    if (use_ws) {
        const int tiles  = n >> 4;           // one wave per 16-row tile
        const int blocks = (tiles + 7) >> 3; // 8 waves (256 threads) per block
        volgru_gemm_wmma<<<blocks, 256, 0, stream>>>(X, Bg, Br, GL, RL, n);
    }
    volgru_scan<<<1, 32, 0, stream>>>(GL, RL, X, rets, Bg, gsw, Br, rsw,
                                      W1, b1, W2, b2,
                                      out_sigma, out_z, out_cand, out_vfin,
                                      n, use_ws);
}